// ContrastiveGNN_32366873543266
// MI455X (gfx1250) — compile-verified
//
#include <hip/hip_runtime.h>
#include <hip/hip_bf16.h>

typedef __attribute__((ext_vector_type(2))) float v2f;
typedef __attribute__((ext_vector_type(8))) float v8f;

#define K_DIM 128
#define LDS_PAD 8   // row stride = M+8 floats -> half-wave bank offset 16, conflict-free

// ---------------------------------------------------------------------------
// Degree / normalization
// ---------------------------------------------------------------------------
__global__ void gcn_deg_init(float* __restrict__ deg, int n) {
    int i = blockIdx.x * blockDim.x + threadIdx.x;
    if (i < n) deg[i] = 1.0f;  // self-loop contributes 1
}

__global__ void gcn_deg_count(const int* __restrict__ dst, float* __restrict__ deg, int nE) {
    int e = blockIdx.x * blockDim.x + threadIdx.x;
    if (e < nE) atomicAdd(&deg[dst[e]], 1.0f);
}

__global__ void gcn_rsqrt(float* __restrict__ d, int n) {
    int i = blockIdx.x * blockDim.x + threadIdx.x;
    if (i < n) d[i] = rsqrtf(fmaxf(d[i], 1.0f));
}

// ---------------------------------------------------------------------------
// GEMM: H[N x M] = (RELU_IN ? relu(X) : X)[N x 128] @ W[128 x M]
// One wave computes a 16-row strip across all M columns using
// V_WMMA_F32_16X16X4_F32. W is staged in LDS with a padded row stride
// (M+8 floats) so the two half-waves of a B-fragment ds_load hit disjoint
// bank groups (offset 2 rows = 2*(M+8) floats; mod 64 banks = 16).
// ---------------------------------------------------------------------------
template <int M, bool RELU_IN>
__global__ void __launch_bounds__(256)
gcn_gemm_wmma(const float* __restrict__ X, const float* __restrict__ W,
              float* __restrict__ H, int nRows) {
    constexpr int LSTR = M + LDS_PAD;
    __shared__ float Wlds[K_DIM * LSTR];

    const int tid = threadIdx.x;
    // cooperative load of W (128 x M) into padded LDS rows, float4 at a time
    for (int i = tid * 4; i < K_DIM * M; i += 256 * 4) {
        const int r = i / M;
        const int c = i - r * M;                 // M multiple of 4 -> c aligned to 4
        *(float4*)&Wlds[r * LSTR + c] = *(const float4*)&W[i];
    }
    __syncthreads();

    const int wave = tid >> 5;
    const int lane = tid & 31;
    const int tile = blockIdx.x * 8 + wave;          // 16-row tile index
    if (tile * 16 >= nRows) return;                  // wave-uniform: EXEC stays all-1

    const int rowA = tile * 16 + (lane & 15);        // A layout: both halves map M=0..15
    const int kOff = (lane >> 4) * 2;                // lanes 0-15 -> K+0/1, lanes 16-31 -> K+2/3
    const int nCol = (lane & 15);
    const float* __restrict__ xrow = X + (size_t)rowA * K_DIM;
    const float* __restrict__ wcol = &Wlds[kOff * LSTR + nCol];

    v8f acc[M / 16] = {};

#pragma unroll 4
    for (int k = 0; k < K_DIM; k += 4) {
        float a0 = xrow[k + kOff];
        float a1 = xrow[k + kOff + 1];
        if (RELU_IN) { a0 = fmaxf(a0, 0.0f); a1 = fmaxf(a1, 0.0f); }
        v2f a; a.x = a0; a.y = a1;
        const float* __restrict__ wk = wcol + k * LSTR;
#pragma unroll
        for (int t = 0; t < M / 16; ++t) {
            v2f b;
            b.x = wk[t * 16];            // row k+kOff
            b.y = wk[LSTR + t * 16];     // row k+kOff+1
            acc[t] = __builtin_amdgcn_wmma_f32_16x16x4_f32(
                false, a, false, b, (short)0, acc[t], false, false);
        }
    }

    // C/D layout: VGPR v -> row v (lanes 0-15) and row v+8 (lanes 16-31)
    const int rBase = tile * 16 + ((lane >> 4) << 3);
#pragma unroll
    for (int t = 0; t < M / 16; ++t) {
#pragma unroll
        for (int v = 0; v < 8; ++v) {
            H[(size_t)(rBase + v) * M + t * 16 + nCol] = acc[t][v];
        }
    }
}

// ---------------------------------------------------------------------------
// Aggregation: out[i] = sum_{e: dst=i} H[src_e]*dis[src]*dis[dst]
//                        + H[i]*dis[i]^2 (self-loop) + bias
// agg_init writes self-loop term + bias (full overwrite -> no memset);
// agg_edges scatters edge contributions with fp32 atomics (L2-resident:
// the 51 MB feature table fits in the 192 MB L2).
// ---------------------------------------------------------------------------
template <int M>
__global__ void gcn_agg_init(const float* __restrict__ H, const float* __restrict__ dis,
                             const float* __restrict__ bias, float* __restrict__ out, int n) {
    int gid = blockIdx.x * blockDim.x + threadIdx.x;
    if (gid >= n * M) return;
    int i = gid / M;
    int j = gid - i * M;
    float s = dis[i];
    out[gid] = H[gid] * (s * s) + bias[j];
}

template <int M>
__global__ void gcn_agg_edges(const float* __restrict__ H,
                              const int* __restrict__ src, const int* __restrict__ dst,
                              const float* __restrict__ dis,
                              float* __restrict__ out, int nE) {
    int gid = blockIdx.x * blockDim.x + threadIdx.x;
    int e    = gid >> 5;
    int lane = gid & 31;
    if (e >= nE) return;
    int s = src[e];
    int d = dst[e];
    float w = dis[s] * dis[d];
    const float* __restrict__ hs = H + (size_t)s * M;
    float* __restrict__ od = out + (size_t)d * M;
#pragma unroll
    for (int j = lane; j < M; j += 32) {
        atomicAdd(&od[j], hs[j] * w);
    }
}

// ---------------------------------------------------------------------------
// Launch
// ---------------------------------------------------------------------------
extern "C" void kernel_launch(void* const* d_in, const int* in_sizes, int n_in,
                              void* d_out, int out_size, void* d_ws, size_t ws_size,
                              hipStream_t stream) {
    (void)n_in; (void)out_size; (void)ws_size;

    const float* x  = (const float*)d_in[0];
    const int*   ei = (const int*)d_in[1];
    const float* W1 = (const float*)d_in[2];
    const float* b1 = (const float*)d_in[3];
    const float* W2 = (const float*)d_in[4];
    const float* b2 = (const float*)d_in[5];
    const float* W3 = (const float*)d_in[6];
    const float* b3 = (const float*)d_in[7];
    float* out = (float*)d_out;

    const int N = in_sizes[0] / K_DIM;        // 100000
    const int E = in_sizes[1] / 2;            // 1600000
    const int* src = ei;
    const int* dst = ei + E;

    // workspace layout: dis[N] | Bh[N*128] | Ba[N*128]
    float* dis = (float*)d_ws;
    float* Bh  = dis + N;                     // N*4 = 400000 bytes, 16B aligned
    float* Ba  = Bh + (size_t)N * 128;

    const int T = 256;
    const int gN    = (N + T - 1) / T;
    const int gE    = (E + T - 1) / T;
    const int gEdge = (int)(((long long)E * 32 + T - 1) / T);
    const int tiles = (N + 15) / 16;          // 6250 (exact for N=100000)
    const int gGemm = (tiles + 7) / 8;

    // normalization: deg (incl. self-loop) -> dis = rsqrt(deg)
    gcn_deg_init<<<gN, T, 0, stream>>>(dis, N);
    gcn_deg_count<<<gE, T, 0, stream>>>(dst, dis, E);
    gcn_rsqrt<<<gN, T, 0, stream>>>(dis, N);

    // layer 1: H = x@W1 ; Ba = aggregate(H) + b1  (ReLU fused into next GEMM read)
    gcn_gemm_wmma<128, false><<<gGemm, T, 0, stream>>>(x, W1, Bh, N);
    gcn_agg_init<128><<<(N * 128 + T - 1) / T, T, 0, stream>>>(Bh, dis, b1, Ba, N);
    gcn_agg_edges<128><<<gEdge, T, 0, stream>>>(Bh, src, dst, dis, Ba, E);

    // layer 2
    gcn_gemm_wmma<128, true><<<gGemm, T, 0, stream>>>(Ba, W2, Bh, N);
    gcn_agg_init<128><<<(N * 128 + T - 1) / T, T, 0, stream>>>(Bh, dis, b2, Ba, N);
    gcn_agg_edges<128><<<gEdge, T, 0, stream>>>(Bh, src, dst, dis, Ba, E);

    // layer 3 (M=64, no ReLU on output) -> d_out
    gcn_gemm_wmma<64, true><<<gGemm, T, 0, stream>>>(Ba, W3, Bh, N);
    gcn_agg_init<64><<<(N * 64 + T - 1) / T, T, 0, stream>>>(Bh, dis, b3, out, N);
    gcn_agg_edges<64><<<gEdge, T, 0, stream>>>(Bh, src, dst, dis, out, E);
}